// Bottom_Right_Corner_66623532695950
// MI455X (gfx1250) — compile-verified
//
#include <hip/hip_runtime.h>
#include <stdint.h>

// Bottom+Right corner pool: out = 2 * cummax_W(cummax_H(x)), x: (16,256,128,128) f32.
// Memory-bound (512 MB total @ 23.3 TB/s ~= 22 us floor). One workgroup per
// 128x128 plane; whole plane staged in LDS via CDNA5 async global<->LDS DMA
// (GLOBAL_LOAD_ASYNC_TO_LDS_B64 / GLOBAL_STORE_ASYNC_FROM_LDS_B64, ASYNCcnt).

#define HH       128
#define WW       128
#define SS       130   // padded LDS row stride (floats): 8B-aligned rows, bank-conflict-free column walks
#define THREADS  128
#define NPLANE   (HH * WW)

// Builtin prototype is (V2i addrspace(1)*, V2i addrspace(3)*, imm int, imm int)
typedef int v2i __attribute__((vector_size(8)));
typedef __attribute__((address_space(1))) v2i* gp64_t;
typedef __attribute__((address_space(3))) v2i* lp64_t;

__global__ __launch_bounds__(THREADS)
void corner_pool_2d_prefix_max_kernel(const float* __restrict__ x,
                                      float* __restrict__ out) {
    __shared__ float tile[HH * SS];

    const int t     = threadIdx.x;
    const int plane = blockIdx.x;
    const float* src = x   + (size_t)plane * NPLANE;
    float*       dst = out + (size_t)plane * NPLANE;

    // ---- Phase 0: async DMA global -> LDS (b64: 8 B/lane, 1 KB/block-iter = 2 rows) ----
    {
        const int h0 = t >> 6;          // which of the 2 rows this iteration
        const int w2 = (t & 63) * 2;    // even column index
        for (int k = 0; k < HH / 2; ++k) {
            const int h = k * 2 + h0;
            const float* g = src + h * WW + w2;
            float*       l = &tile[h * SS + w2];
            __builtin_amdgcn_global_load_async_to_lds_b64(
                (gp64_t)g, (lp64_t)l, /*offset=*/0, /*cpol=*/0);
        }
        __builtin_amdgcn_s_wait_asynccnt(0);
    }
    __syncthreads();

    // ---- Phase 1: cummax down H, in place (thread = column; serial dep along h) ----
    {
        const int w = t;
        float m = tile[w];                       // h = 0 row unchanged
        for (int h = 1; h < HH; ++h) {
            const float v = tile[h * SS + w];
            m = fmaxf(m, v);
            tile[h * SS + w] = m;
        }
    }
    __syncthreads();

    // ---- Phase 2: cummax along W, fold in the *2, in place (thread = row) ----
    // Bank index = (2h + w) mod 64: distinct across 32 consecutive lanes -> conflict-free.
    {
        const int h = t;
        float m = tile[h * SS];
        tile[h * SS] = m + m;
        for (int w = 1; w < WW; ++w) {
            const float v = tile[h * SS + w];
            m = fmaxf(m, v);
            tile[h * SS + w] = m + m;
        }
    }
    __syncthreads();

    // ---- Phase 3: async DMA LDS -> global ----
    {
        const int h0 = t >> 6;
        const int w2 = (t & 63) * 2;
        for (int k = 0; k < HH / 2; ++k) {
            const int h = k * 2 + h0;
            float* g = dst + h * WW + w2;
            float* l = &tile[h * SS + w2];
            __builtin_amdgcn_global_store_async_from_lds_b64(
                (gp64_t)g, (lp64_t)l, /*offset=*/0, /*cpol=*/0);
        }
        __builtin_amdgcn_s_wait_asynccnt(0);   // S_ENDPGM also implies wait-idle
    }
}

extern "C" void kernel_launch(void* const* d_in, const int* in_sizes, int n_in,
                              void* d_out, int out_size, void* d_ws, size_t ws_size,
                              hipStream_t stream) {
    (void)n_in; (void)d_ws; (void)ws_size; (void)out_size;
    const float* x   = (const float*)d_in[0];
    float*       out = (float*)d_out;
    const int planes = in_sizes[0] / NPLANE;   // 16*256 = 4096
    corner_pool_2d_prefix_max_kernel<<<planes, THREADS, 0, stream>>>(x, out);
}